// GCN_85529978733009
// MI455X (gfx1250) — compile-verified
//
#include <hip/hip_runtime.h>

typedef float v2f __attribute__((ext_vector_type(2)));
typedef float v8f __attribute__((ext_vector_type(8)));

#define GCN_N 100000
#define GCN_E 3200000
#define GCN_H 128
#define GCN_O 64

// ---------------------------------------------------------------------------
// deg[n] = 1.0 (self loop weight)
__global__ __launch_bounds__(256) void k_init_deg(float* __restrict__ deg, int n) {
    int i = blockIdx.x * 256 + threadIdx.x;
    if (i < n) deg[i] = 1.0f;
}

// deg[dst] += 1 for each non-self edge
__global__ __launch_bounds__(256) void k_count_deg(const int* __restrict__ src,
                                                   const int* __restrict__ dst,
                                                   float* __restrict__ deg, int e) {
    int i = blockIdx.x * 256 + threadIdx.x;
    if (i >= e) return;
    int s = src[i], d = dst[i];
    if (s != d) atomicAdd(&deg[d], 1.0f);
}

// dinv = rsqrt(deg)  (deg >= 1 always)
__global__ __launch_bounds__(256) void k_rsqrt(float* __restrict__ deg, int n) {
    int i = blockIdx.x * 256 + threadIdx.x;
    if (i < n) deg[i] = rsqrtf(deg[i]);
}

// agg[n][:] = bias[:] + dinv[n]^2 * hw[n][:]   (self-loop + layer bias folded in)
__global__ __launch_bounds__(256) void k_init_agg(const float* __restrict__ dinv,
                                                  const float* __restrict__ hw,
                                                  const float* __restrict__ bias,
                                                  float* __restrict__ agg) {
    long long gid = (long long)blockIdx.x * 256 + threadIdx.x;   // N*32 threads
    int n = (int)(gid >> 5);
    int c = ((int)gid & 31) << 2;
    if (n >= GCN_N) return;
    float di = dinv[n];
    float w = di * di;
    float4 v = *reinterpret_cast<const float4*>(&hw[(size_t)n * GCN_H + c]);
    float4 b = *reinterpret_cast<const float4*>(&bias[c]);
    float4 o;
    o.x = b.x + w * v.x; o.y = b.y + w * v.y;
    o.z = b.z + w * v.z; o.w = b.w + w * v.w;
    *reinterpret_cast<float4*>(&agg[(size_t)n * GCN_H + c]) = o;
}

// agg[dst][:] += dinv[src]*dinv[dst] * hw[src][:]   for src != dst
// one wave per edge: 32 lanes x float4 = 128 floats, coalesced L2-resident gather
__global__ __launch_bounds__(256) void k_scatter(const int* __restrict__ src,
                                                 const int* __restrict__ dst,
                                                 const float* __restrict__ dinv,
                                                 const float* __restrict__ hw,
                                                 float* __restrict__ agg, int e) {
    long long gid = (long long)blockIdx.x * 256 + threadIdx.x;   // E*32 threads
    int ei = (int)(gid >> 5);
    int c = ((int)gid & 31) << 2;
    if (ei >= e) return;
    int s = src[ei], d = dst[ei];
    if (s == d) return;                               // dropped self-loop (weight 0)
    float nw = dinv[s] * dinv[d];
    float4 v = *reinterpret_cast<const float4*>(&hw[(size_t)s * GCN_H + c]);
    float* o = &agg[(size_t)d * GCN_H + c];
    atomicAdd(o + 0, nw * v.x);
    atomicAdd(o + 1, nw * v.y);
    atomicAdd(o + 2, nw * v.z);
    atomicAdd(o + 3, nw * v.w);
}

// ---------------------------------------------------------------------------
// Prepack B[128 x NCOLS] into per-lane WMMA fragment order:
// Bp[(ct*32 + i)*32 + lane] = { B[(4i+kb)*NCOLS + ct*16 + m],
//                              B[(4i+kb+1)*NCOLS + ct*16 + m] }
// where m = lane&15, kb = (lane>>4)*2.  In the GEMM loop each lane then loads
// a single b64 at base + i*256B (pure immediate offsets, no address math).
template <int NCOLS>
__global__ __launch_bounds__(256) void k_pack_b(const float* __restrict__ B,
                                                float* __restrict__ Bp) {
    constexpr int colTiles = NCOLS / 16;
    int t = blockIdx.x * 256 + threadIdx.x;          // colTiles*32*32 threads
    if (t >= colTiles * 1024) return;
    int lane = t & 31;
    int i = (t >> 5) & 31;
    int ct = t >> 10;
    int m = lane & 15;
    int kb = (lane >> 4) << 1;
    int k = 4 * i + kb;
    float2 v;
    v.x = B[k * NCOLS + ct * 16 + m];
    v.y = B[(k + 1) * NCOLS + ct * 16 + m];
    reinterpret_cast<float2*>(Bp)[t] = v;
}

// ---------------------------------------------------------------------------
// C[M x NCOLS] = op(A[M x 128]) @ B (prepacked), fp32 WMMA 16x16x4.
// RELU_A: ReLU fused into LDS staging of A.
// MODE 0: store acc; 1: relu((acc+bias)*gamma/sqrt(1+eps)+beta); 2: acc+bias
template <int NCOLS, int RELU_A, int MODE>
__global__ __launch_bounds__(256) void k_gemm_wmma(const float* __restrict__ A,
                                                   const float* __restrict__ Bp,
                                                   float* __restrict__ C, int M,
                                                   const float* __restrict__ bias,
                                                   const float* __restrict__ gamma,
                                                   const float* __restrict__ beta) {
    constexpr int K = 128;
    constexpr int LDA = 132;                     // +4 pad: conflict-free ds_load_b64
    constexpr int colTiles = NCOLS / 16;         // 8 or 4
    constexpr int rowTilesPerBlock = 8 / colTiles;
    constexpr int rowsPerBlock = rowTilesPerBlock * 16;
    __shared__ float As[rowsPerBlock * LDA];

    int blockRow0 = blockIdx.x * rowsPerBlock;

    // --- stage A tile (rowsPerBlock x 128) into LDS, fused ReLU ---
    {
        constexpr int elems = rowsPerBlock * K;  // 2048 or 4096
#pragma unroll
        for (int i = threadIdx.x * 4; i < elems; i += 256 * 4) {
            int r = i >> 7;
            int c = i & 127;
            float4 v = *reinterpret_cast<const float4*>(
                &A[(size_t)(blockRow0 + r) * K + c]);
            if (RELU_A) {
                v.x = fmaxf(v.x, 0.f); v.y = fmaxf(v.y, 0.f);
                v.z = fmaxf(v.z, 0.f); v.w = fmaxf(v.w, 0.f);
            }
            *reinterpret_cast<float4*>(&As[r * LDA + c]) = v;
        }
    }
    __syncthreads();

    int wave = threadIdx.x >> 5;
    int lane = threadIdx.x & 31;
    int rt = wave / colTiles;
    int ct = wave - rt * colTiles;
    int m  = lane & 15;                          // M (A) / N (B,C) within tile
    int kb = (lane >> 4) << 1;                   // 0 or 2: K-pair per lane half
    int bn = ct << 4;

    const float* Arow = &As[(rt * 16 + m) * LDA];
    const float2* Bw = reinterpret_cast<const float2*>(Bp) + (ct * 1024 + lane);

    v8f acc = {};
#pragma unroll
    for (int i = 0; i < 32; ++i) {
        float2 bf = Bw[i * 32];                  // b64 load, immediate offset
        v2f a, b;
        a.x = Arow[4 * i + kb];                  // ds_load_2addr_b64 pair
        a.y = Arow[4 * i + kb + 1];
        b.x = bf.x;
        b.y = bf.y;
        acc = __builtin_amdgcn_wmma_f32_16x16x4_f32(
            /*neg_a=*/false, a, /*neg_b=*/false, b,
            /*c_mod=*/(short)0, acc, /*reuse_a=*/false, /*reuse_b=*/false);
    }

    // --- epilogue ---
    int col = bn + m;
    float bval = (MODE != 0) ? bias[col] : 0.f;
    float g = 1.f, bt = 0.f;
    if (MODE == 1) {
        g = gamma[col] * rsqrtf(1.0f + 1e-5f);   // BN eval: mean=0, var=1
        bt = beta[col];
    }
    int rbase = blockRow0 + rt * 16 + ((lane >> 4) << 3);  // upper half -> M+8
    float* Cp = &C[(size_t)rbase * NCOLS + col];
#pragma unroll
    for (int v = 0; v < 8; ++v) {
        float val = acc[v];
        if (MODE != 0) val += bval;
        if (MODE == 1) { val = val * g + bt; val = fmaxf(val, 0.f); }
        if (rbase + v < M) Cp[v * NCOLS] = val;  // immediate-offset stores
    }
}

// ---------------------------------------------------------------------------
extern "C" void kernel_launch(void* const* d_in, const int* in_sizes, int n_in,
                              void* d_out, int out_size, void* d_ws, size_t ws_size,
                              hipStream_t stream) {
    const int N = GCN_N, E = GCN_E, H = GCN_H;

    const float* x     = (const float*)d_in[0];
    const int*   edge  = (const int*)d_in[1];     // [2, E]
    const float* Ws    = (const float*)d_in[2];   // [3, 128, 128]
    const float* bs    = (const float*)d_in[3];   // [3, 128]
    const float* wo1   = (const float*)d_in[4];   // [128, 128]
    const float* bo1   = (const float*)d_in[5];
    const float* gamma = (const float*)d_in[6];
    const float* beta  = (const float*)d_in[7];
    const float* wo2   = (const float*)d_in[8];   // [128, 64]
    const float* bo2   = (const float*)d_in[9];
    float* out = (float*)d_out;

    const int* esrc = edge;
    const int* edst = edge + E;

    float* bufA  = (float*)d_ws;                  // [N,128]
    float* bufB  = bufA + (size_t)N * H;          // [N,128]
    float* dinv  = bufB + (size_t)N * H;          // [N] (deg -> dinv in place)
    float* bpack = dinv + N;                      // [128*128] packed B fragments

    // degree normalization
    k_init_deg<<<(N + 255) / 256, 256, 0, stream>>>(dinv, N);
    k_count_deg<<<(E + 255) / 256, 256, 0, stream>>>(esrc, edst, dinv, E);
    k_rsqrt<<<(N + 255) / 256, 256, 0, stream>>>(dinv, N);

    const int gemm128_grid = N / 16;              // 6250 (exact)
    const int gemm64_grid  = N / 32;              // 3125 (exact)
    const int agg_grid     = (N * 32 + 255) / 256;
    const long long sc_threads = (long long)E * 32;
    const int sc_grid = (int)((sc_threads + 255) / 256);

    // 3 GCN layers: h = relu(segsum(norm * (h @ W)) + b)
    for (int l = 0; l < 3; ++l) {
        k_pack_b<128><<<32, 256, 0, stream>>>(Ws + (size_t)l * H * H, bpack);
        if (l == 0)
            k_gemm_wmma<128, 0, 0><<<gemm128_grid, 256, 0, stream>>>(
                x, bpack, bufB, N, nullptr, nullptr, nullptr);
        else
            k_gemm_wmma<128, 1, 0><<<gemm128_grid, 256, 0, stream>>>(
                bufA, bpack, bufB, N, nullptr, nullptr, nullptr);
        k_init_agg<<<agg_grid, 256, 0, stream>>>(dinv, bufB, bs + (size_t)l * H, bufA);
        k_scatter<<<sc_grid, 256, 0, stream>>>(esrc, edst, dinv, bufB, bufA, E);
    }

    // head: relu(BN(h @ wo1 + bo1)) @ wo2 + bo2
    k_pack_b<128><<<32, 256, 0, stream>>>(wo1, bpack);
    k_gemm_wmma<128, 1, 1><<<gemm128_grid, 256, 0, stream>>>(
        bufA, bpack, bufB, N, bo1, gamma, beta);
    k_pack_b<64><<<16, 256, 0, stream>>>(wo2, bpack);
    k_gemm_wmma<64, 0, 2><<<gemm64_grid, 256, 0, stream>>>(
        bufB, bpack, out, N, bo2, nullptr, nullptr);
}